// FisherLayer_30846455119907
// MI455X (gfx1250) — compile-verified
//
#include <hip/hip_runtime.h>
#include <math.h>

typedef float v2f __attribute__((ext_vector_type(2)));
typedef float v8f __attribute__((ext_vector_type(8)));

#define NROWS 8192
#define DDIM  256
#define KK    32

// workspace layout (float offsets)
#define WS_WT         0        // W' transposed, [32][512]
#define WS_CC         16384    // C[k], 32
#define WS_G0         16416    // 32
#define WS_G1         16448    // [32][256]
#define WS_G2         24640    // [32][256]
#define WS_ZERO_BEGIN 16416
#define WS_ZERO_LEN   16416

static __device__ __forceinline__ v8f wmma_f32(v2f a, v2f b, v8f c) {
  // D = A(16x4 f32) x B(4x16 f32) + C(16x16 f32)
  return __builtin_amdgcn_wmma_f32_16x16x4_f32(false, a, false, b, (short)0, c,
                                               false, false);
}

// ---------------------------------------------------------------------------
// prep: zero accumulators, build W' = [w^2 ; 2*w^2*b] (k-major) and C[k]
// ---------------------------------------------------------------------------
__global__ __launch_bounds__(256) void fisher_prep(const float* __restrict__ w,
                                                   const float* __restrict__ b,
                                                   float* __restrict__ ws) {
  int tid = threadIdx.x;
  for (int i = tid; i < WS_ZERO_LEN; i += 256) ws[WS_ZERO_BEGIN + i] = 0.0f;
  float* Wt = ws + WS_WT;
  for (int i = tid; i < KK * DDIM; i += 256) {
    int k = i >> 8, d = i & 255;
    float wv = w[i];
    float bv = b[i];
    float w2 = wv * wv;
    Wt[k * 512 + d]       = w2;
    Wt[k * 512 + 256 + d] = 2.0f * w2 * bv;
  }
  if (tid < KK) {
    float c = 0.0f;
    for (int d = 0; d < DDIM; ++d) {
      float wv = w[tid * DDIM + d];
      float bv = b[tid * DDIM + d];
      c += wv * wv * bv * bv;
    }
    ws[WS_CC + tid] = c;
  }
}

// ---------------------------------------------------------------------------
// fused: pass1 (logit GEMM + softmax, gamma -> LDS) then pass2 (gamma^T GEMM)
// 64 blocks x 128 threads (4 waves); 128 rows per block.
// ---------------------------------------------------------------------------
__global__ __launch_bounds__(128) void fisher_fused(const float* __restrict__ x,
                                                    float* __restrict__ ws) {
  __shared__ float gamma_lds[128 * 32];   // [local_row][k], 16 KB

  const float* Wt = ws + WS_WT;
  const float* Cc = ws + WS_CC;
  float* G0 = ws + WS_G0;
  float* G1 = ws + WS_G1;
  float* G2 = ws + WS_G2;

  const int lane = threadIdx.x & 31;
  const int wv   = threadIdx.x >> 5;
  const int half = lane >> 4;    // 0: lanes 0-15, 1: lanes 16-31
  const int l15  = lane & 15;
  const int row_block = blockIdx.x * 128;
  const int kA = 2 * half;       // K-offset of this lane-half in A/B operands

  float g0sum = 0.0f, g1sum = 0.0f;

  // ------------------- pass 1: Y4 = [x^2,x] @ W' , softmax over k ----------
  for (int t = wv * 2; t < wv * 2 + 2; ++t) {
    const int row0 = row_block + t * 16;
    const float* xrow = x + (size_t)(row0 + l15) * DDIM;   // A: M = l15
    const float* wt0  = Wt + l15 * 512;                    // B: N = l15  (k)
    const float* wt1  = Wt + (l15 + 16) * 512;             // B: k + 16

    v8f acc0 = {};
    v8f acc1 = {};
    // features 0..255: x^2  (branch-free straight-line loop)
#pragma unroll 4
    for (int f0 = 0; f0 < 256; f0 += 4) {
      int fb = f0 + kA;
      v2f u  = *(const v2f*)(xrow + fb);
      v2f a  = u * u;
      v2f b0 = *(const v2f*)(wt0 + fb);
      v2f b1 = *(const v2f*)(wt1 + fb);
      acc0 = wmma_f32(a, b0, acc0);
      acc1 = wmma_f32(a, b1, acc1);
    }
    // features 256..511: x
#pragma unroll 4
    for (int f0 = 0; f0 < 256; f0 += 4) {
      int fb = f0 + kA;
      v2f a  = *(const v2f*)(xrow + fb);
      v2f b0 = *(const v2f*)(wt0 + 256 + fb);
      v2f b1 = *(const v2f*)(wt1 + 256 + fb);
      acc0 = wmma_f32(a, b0, acc0);
      acc1 = wmma_f32(a, b1, acc1);
    }

    const float Ck0 = Cc[l15];
    const float Ck1 = Cc[l15 + 16];
#pragma unroll
    for (int r = 0; r < 8; ++r) {
      // lane holds (row = r + 8*half, k = l15 [+16]) of the 16x32 tile
      float y0 = -0.5f * (acc0[r] + Ck0);
      float y1 = -0.5f * (acc1[r] + Ck1);
      // row max across the 16 lanes of this half (xor 1,2,4,8 stays in-half)
      float m = fmaxf(y0, y1);
      m = fmaxf(m, __shfl_xor(m, 1, 32));
      m = fmaxf(m, __shfl_xor(m, 2, 32));
      m = fmaxf(m, __shfl_xor(m, 4, 32));
      m = fmaxf(m, __shfl_xor(m, 8, 32));
      float e0 = __expf(y0 - m);
      float e1 = __expf(y1 - m);
      float s = e0 + e1;
      s += __shfl_xor(s, 1, 32);
      s += __shfl_xor(s, 2, 32);
      s += __shfl_xor(s, 4, 32);
      s += __shfl_xor(s, 8, 32);
      float inv = 1.0f / s;
      float g0 = e0 * inv;
      float g1 = e1 * inv;
      int mrow = t * 16 + r + 8 * half;     // local row 0..127
      gamma_lds[mrow * 32 + l15]      = g0;
      gamma_lds[mrow * 32 + l15 + 16] = g1;
      g0sum += g0;
      g1sum += g1;
    }
  }
  atomicAdd(&G0[l15],      g0sum);
  atomicAdd(&G0[l15 + 16], g1sum);
  __syncthreads();

  // ------------------- pass 2: G1 = gamma^T @ x, G2 = gamma^T @ x^2 --------
  for (int dt = wv * 4; dt < wv * 4 + 4; ++dt) {
    const int d = dt * 16 + l15;            // B: N = d
    v8f a10 = {}, a20 = {}, a11 = {}, a21 = {};
#pragma unroll 2
    for (int ns = 0; ns < 32; ++ns) {
      int nloc = ns * 4 + kA;               // A/B: K = local row index
      // A operand: M = k, K = n  -> gamma_lds[n][k]
      v2f A0 = { gamma_lds[nloc * 32 + l15],      gamma_lds[(nloc + 1) * 32 + l15] };
      v2f A1 = { gamma_lds[nloc * 32 + l15 + 16], gamma_lds[(nloc + 1) * 32 + l15 + 16] };
      float x0 = x[(size_t)(row_block + nloc) * DDIM + d];
      float x1 = x[(size_t)(row_block + nloc + 1) * DDIM + d];
      v2f B1v = { x0, x1 };
      v2f B2v = { x0 * x0, x1 * x1 };
      a10 = wmma_f32(A0, B1v, a10);
      a20 = wmma_f32(A0, B2v, a20);
      a11 = wmma_f32(A1, B1v, a11);
      a21 = wmma_f32(A1, B2v, a21);
    }
#pragma unroll
    for (int r = 0; r < 8; ++r) {
      int krow = r + 8 * half;              // C/D: M = k row
      atomicAdd(&G1[krow * 256 + d],        a10[r]);
      atomicAdd(&G2[krow * 256 + d],        a20[r]);
      atomicAdd(&G1[(krow + 16) * 256 + d], a11[r]);
      atomicAdd(&G2[(krow + 16) * 256 + d], a21[r]);
    }
  }
}

// ---------------------------------------------------------------------------
// final: epilogue over 2*K*D outputs
// ---------------------------------------------------------------------------
__global__ __launch_bounds__(256) void fisher_final(const float* __restrict__ w,
                                                    const float* __restrict__ b,
                                                    const float* __restrict__ ws,
                                                    float* __restrict__ out) {
  int idx = blockIdx.x * 256 + threadIdx.x;  // 0..8191  (k*256 + d)
  if (idx >= KK * DDIM) return;
  int k = idx >> 8;
  float g0 = ws[WS_G0 + k];
  float g1 = ws[WS_G1 + idx];
  float g2 = ws[WS_G2 + idx];
  float wv = w[idx], bv = b[idx];
  float w2 = wv * wv;
  float S2 = w2 * (g2 + 2.0f * bv * g1 + bv * bv * g0);   // sum_n gamma*y1^2
  const float invN = 1.0f / 8192.0f;
  const float invs2 = 0.70710678118654752440f;            // 1/sqrt(2)
  out[idx]             = (S2 - g0) * invN * invs2;        // sigma part
  out[KK * DDIM + idx] = wv * (g1 + bv * g0) * invN;      // mu part
}

// ---------------------------------------------------------------------------
extern "C" void kernel_launch(void* const* d_in, const int* in_sizes, int n_in,
                              void* d_out, int out_size, void* d_ws, size_t ws_size,
                              hipStream_t stream) {
  (void)in_sizes; (void)n_in; (void)out_size; (void)ws_size;
  const float* x = (const float*)d_in[0];
  const float* w = (const float*)d_in[1];
  const float* b = (const float*)d_in[2];
  float* out = (float*)d_out;
  float* ws  = (float*)d_ws;   // needs ~128.3 KB of float scratch

  fisher_prep <<<1, 256, 0, stream>>>(w, b, ws);
  fisher_fused<<<NROWS / 128, 128, 0, stream>>>(x, ws);
  fisher_final<<<(KK * DDIM + 255) / 256, 256, 0, stream>>>(w, b, ws, out);
}